// quantumAttentionBlock_81922206203910
// MI455X (gfx1250) — compile-verified
//
#include <hip/hip_runtime.h>
#include <math.h>

// ---------------------------------------------------------------------------
// Quantum attention block for MI455X (gfx1250), wave32, f32 WMMA 16x16x4.
// B=32, N=1024, D=16, H=64.  Everything fits in L2 (192MB); the NxN attention
// matrix lives only in LDS (16x1028 f32 per row-tile = ~66KB of 320KB/WGP).
// Fast hardware transcendentals (v_sqrt/v_rsq/v_rcp, ~1ulp) replace the IEEE
// fixup sequences in the hot loops.
// ---------------------------------------------------------------------------

#define BATCH 32
#define NTOK  1024
#define DDIM  16
#define HDIM  64
#define SROW  1028   // padded LDS row stride in dwords (1028 % 64 == 4 -> lanes
                     // 0..15 map to 16 distinct banks, halves interleave: 32 banks)

typedef __attribute__((ext_vector_type(2))) float v2f;
typedef __attribute__((ext_vector_type(8))) float v8f;

__device__ __forceinline__ float fast_sqrt(float x)  { return __builtin_amdgcn_sqrtf(x); }
__device__ __forceinline__ float fast_rsqrt(float x) { return __builtin_amdgcn_rsqf(x); }
__device__ __forceinline__ float fast_rcp(float x)   { return __builtin_amdgcn_rcpf(x); }

// ---------------------------------------------------------------------------
// Kernel 0: zero the per-(b,d) sum-of-squares accumulator (ws is poisoned).
// ---------------------------------------------------------------------------
__global__ void qab_zero_kernel(float* __restrict__ p, int n) {
    int i = blockIdx.x * 256 + threadIdx.x;
    if (i < n) p[i] = 0.0f;
}

// ---------------------------------------------------------------------------
// Kernel 1: LayerNorm over D=16 + partial sum of h^2 over the token axis.
// Grid: (N/256, B) x 256 threads; one token per thread.
// ---------------------------------------------------------------------------
__global__ __launch_bounds__(256)
void qab_ln1_kernel(const float* __restrict__ x,
                    const float* __restrict__ w,
                    const float* __restrict__ bia,
                    float* __restrict__ hbuf,
                    float* __restrict__ sumsq) {
    __shared__ float part[256][DDIM + 1];   // +1 pad: spread banks
    const int b   = blockIdx.y;
    const int tok = blockIdx.x * 256 + threadIdx.x;          // 0..1023
    const size_t base = ((size_t)b * NTOK + tok) * DDIM;
    const float* xr = x + base;

    float v[DDIM];
    float mu = 0.0f;
#pragma unroll
    for (int d = 0; d < DDIM; ++d) { v[d] = xr[d]; mu += v[d]; }
    mu *= (1.0f / DDIM);
    float var = 0.0f;
#pragma unroll
    for (int d = 0; d < DDIM; ++d) { float t = v[d] - mu; var += t * t; }
    var *= (1.0f / DDIM);
    const float inv = fast_rsqrt(var + 1e-5f);

    float* hr = hbuf + base;
#pragma unroll
    for (int d = 0; d < DDIM; ++d) {
        float h = (v[d] - mu) * inv * w[d] + bia[d];
        hr[d] = h;
        part[threadIdx.x][d] = h * h;
    }
    __syncthreads();
    // tree reduce 256 tokens -> 16 partial sums
    for (int off = 128; off > 0; off >>= 1) {
        if (threadIdx.x < (unsigned)off) {
#pragma unroll
            for (int d = 0; d < DDIM; ++d)
                part[threadIdx.x][d] += part[threadIdx.x + off][d];
        }
        __syncthreads();
    }
    if (threadIdx.x < DDIM)
        atomicAdd(&sumsq[b * DDIM + threadIdx.x], part[0][threadIdx.x]);
}

// ---------------------------------------------------------------------------
// Kernel 2: token-axis normalization + projections hW and sqrt((Vh)^2+eps).
// One token per thread; W,V are uniform -> scalar-cached loads.
// ---------------------------------------------------------------------------
__global__ __launch_bounds__(256)
void qab_proj_kernel(const float* __restrict__ Vm,
                     const float* __restrict__ Wm,
                     const float* __restrict__ sumsq,
                     float* __restrict__ hbuf,
                     float* __restrict__ hwbuf,
                     float* __restrict__ vxsbuf) {
    const int gid = blockIdx.x * 256 + threadIdx.x;          // 0..B*N-1
    const int b   = gid >> 10;
    const float* ss = sumsq + b * DDIM;
    float* hr = hbuf + (size_t)gid * DDIM;

    float h[DDIM];
#pragma unroll
    for (int d = 0; d < DDIM; ++d) {
        // denom = sqrt( sum_n(h^2 + 1e-4) + 1e-8 ) = sqrt( sumsq + N*1e-4 + 1e-8 )
        h[d] = hr[d] * fast_rsqrt(ss[d] + (float)NTOK * 1e-4f + 1e-8f);
    }
#pragma unroll
    for (int d = 0; d < DDIM; ++d) hr[d] = h[d];             // in-place normalize

    float* hwr = hwbuf + (size_t)gid * DDIM;
#pragma unroll
    for (int e = 0; e < DDIM; ++e) {
        float a = 0.0f;
#pragma unroll
        for (int d = 0; d < DDIM; ++d) a += h[d] * Wm[d * DDIM + e];
        hwr[e] = a;                                          // (hW)[e]
    }
    float* vr = vxsbuf + (size_t)gid * DDIM;
#pragma unroll
    for (int d = 0; d < DDIM; ++d) {
        float a = 0.0f;
#pragma unroll
        for (int e = 0; e < DDIM; ++e) a += Vm[d * DDIM + e] * h[e];
        vr[d] = fast_sqrt(a * a + 1e-8f);                    // sqrt((Vh)^2 + eps)
    }
}

// ---------------------------------------------------------------------------
// Kernel 3: fused attention. One wave32 per (batch, 16-row tile) -> 2048 waves.
//   Pass 1: S-tile = A(hW) x B(h^T) via 4 chained V_WMMA_F32_16X16X4_F32,
//           s = sqrt(S^2+1e-8) staged to LDS [16][SROW].
//   Pass 2: row max/sum with float4 LDS scans (ds_load_b128) + shfl_xor 16,
//           exponentiate in place.
//   Pass 3: t = P x vxs via WMMA (A from LDS), *rcp(rowsum), + residual.
// Fragment layouts per ISA 7.12.2 (32-bit A 16x4, mirrored B, 8-VGPR C/D).
// ---------------------------------------------------------------------------
__global__ __launch_bounds__(32)
void qab_attn_kernel(const float* __restrict__ x,
                     const float* __restrict__ hbuf,
                     const float* __restrict__ hwbuf,
                     const float* __restrict__ vxs,
                     float* __restrict__ out) {
    extern __shared__ float smem[];          // [16][SROW] + rowsum[16]
    float* srows  = smem;
    float* rowsum = smem + 16 * SROW;

    const int tile = blockIdx.x;             // 0..2047
    const int b    = tile >> 6;
    const int n0   = (tile & 63) << 4;
    const int lane = threadIdx.x;
    const int half = lane >> 4;              // 0: lanes 0-15, 1: lanes 16-31
    const int l16  = lane & 15;

    // A fragments (hW rows of this n-tile): invariant across column tiles.
    const float* hwrow = hwbuf + ((size_t)(b * NTOK + n0 + l16)) * DDIM;
    v2f afrag[4];
#pragma unroll
    for (int k = 0; k < 4; ++k) {
        afrag[k].x = hwrow[4 * k + 2 * half];
        afrag[k].y = hwrow[4 * k + 2 * half + 1];
    }

    const float* hb = hbuf + (size_t)b * NTOK * DDIM;

    // ---- Pass 1: scores into LDS --------------------------------------
    for (int j = 0; j < 64; ++j) {
        const float* hrow = hb + (size_t)(j * 16 + l16) * DDIM;
        v8f acc = {};
#pragma unroll
        for (int k = 0; k < 4; ++k) {
            v2f bfrag;
            bfrag.x = hrow[4 * k + 2 * half];
            bfrag.y = hrow[4 * k + 2 * half + 1];
            acc = __builtin_amdgcn_wmma_f32_16x16x4_f32(
                false, afrag[k], false, bfrag, (short)0, acc, false, false);
        }
#pragma unroll
        for (int r = 0; r < 8; ++r) {
            int row = r + 8 * half;
            float sv = acc[r];
            sv = fast_sqrt(sv * sv + 1e-8f); // sqrt(xwx^2 + 1e-8)
            srows[row * SROW + j * 16 + l16] = sv;
        }
    }
    __syncthreads();

    // ---- Pass 2: softmax stats + exponentiate (float4 LDS traffic) ----
    {
        const int row = l16;                 // each half-wave scans 512 cols
        float4* rp4 = reinterpret_cast<float4*>(srows + row * SROW + half * 512);
        float mx = -3.4e38f;
        for (int c = 0; c < 128; ++c) {
            float4 q = rp4[c];
            mx = fmaxf(mx, fmaxf(fmaxf(q.x, q.y), fmaxf(q.z, q.w)));
        }
        mx = fmaxf(mx, __shfl_xor(mx, 16, 32));
        float sum = 0.0f;
        for (int c = 0; c < 128; ++c) {
            float4 q = rp4[c];
            q.x = __expf(q.x - mx);
            q.y = __expf(q.y - mx);
            q.z = __expf(q.z - mx);
            q.w = __expf(q.w - mx);
            rp4[c] = q;
            sum += (q.x + q.y) + (q.z + q.w);
        }
        sum += __shfl_xor(sum, 16, 32);
        if (half == 0) rowsum[row] = sum;
    }
    __syncthreads();

    // ---- Pass 3: t = P @ vxs ------------------------------------------
    v8f acc2 = {};
    for (int j = 0; j < 64; ++j) {
        v2f pa[4];
#pragma unroll
        for (int k = 0; k < 4; ++k) {        // A from LDS (conflict-free: pad)
            const float* pp = srows + l16 * SROW + j * 16 + 4 * k + 2 * half;
            pa[k].x = pp[0];
            pa[k].y = pp[1];
        }
#pragma unroll
        for (int k = 0; k < 4; ++k) {
            const int mrow = j * 16 + 4 * k + 2 * half;
            v2f bfrag;
            bfrag.x = vxs[((size_t)(b * NTOK) + mrow) * DDIM + l16];
            bfrag.y = vxs[((size_t)(b * NTOK) + mrow + 1) * DDIM + l16];
            acc2 = __builtin_amdgcn_wmma_f32_16x16x4_f32(
                false, pa[k], false, bfrag, (short)0, acc2, false, false);
        }
    }

    // ---- epilogue: *rcp(rowsum) + residual ----------------------------
#pragma unroll
    for (int r = 0; r < 8; ++r) {
        int row = r + 8 * half;
        float rinv = fast_rcp(rowsum[row]);
        size_t idx = ((size_t)(b * NTOK + n0 + row)) * DDIM + l16;
        out[idx] = x[idx] + acc2[r] * rinv;  // 16 lanes -> 64B coalesced rows
    }
}

// ---------------------------------------------------------------------------
// Kernel 4: LN2 + MLP (gelu tanh approx) + residual, in place on d_out.
// 0.13 GFLOP total -> VALU; weights are uniform (scalar cache).
// ---------------------------------------------------------------------------
__global__ __launch_bounds__(256)
void qab_mlp_kernel(const float* __restrict__ ln2w,
                    const float* __restrict__ ln2b,
                    const float* __restrict__ W1,
                    const float* __restrict__ b1,
                    const float* __restrict__ W2,
                    const float* __restrict__ b2,
                    float* __restrict__ out) {
    const int gid = blockIdx.x * 256 + threadIdx.x;          // token id
    float* xr = out + (size_t)gid * DDIM;

    float v[DDIM];
    float mu = 0.0f;
#pragma unroll
    for (int d = 0; d < DDIM; ++d) { v[d] = xr[d]; mu += v[d]; }
    mu *= (1.0f / DDIM);
    float var = 0.0f;
#pragma unroll
    for (int d = 0; d < DDIM; ++d) { float t = v[d] - mu; var += t * t; }
    var *= (1.0f / DDIM);
    const float inv = fast_rsqrt(var + 1e-5f);

    float g[DDIM], o[DDIM];
#pragma unroll
    for (int d = 0; d < DDIM; ++d) {
        g[d] = (v[d] - mu) * inv * ln2w[d] + ln2b[d];
        o[d] = 0.0f;
    }
    for (int hh = 0; hh < HDIM; ++hh) {
        float a = b1[hh];
#pragma unroll
        for (int d = 0; d < DDIM; ++d) a += g[d] * W1[d * HDIM + hh];
        // jax.nn.gelu default: tanh approximation
        float u = 0.5f * a * (1.0f + tanhf(0.7978845608f * (a + 0.044715f * a * a * a)));
#pragma unroll
        for (int d = 0; d < DDIM; ++d) o[d] += u * W2[hh * DDIM + d];
    }
#pragma unroll
    for (int d = 0; d < DDIM; ++d) xr[d] = v[d] + o[d] + b2[d];
}

// ---------------------------------------------------------------------------
extern "C" void kernel_launch(void* const* d_in, const int* in_sizes, int n_in,
                              void* d_out, int out_size, void* d_ws, size_t ws_size,
                              hipStream_t stream) {
    const float* x     = (const float*)d_in[0];
    const float* ln1_w = (const float*)d_in[1];
    const float* ln1_b = (const float*)d_in[2];
    const float* Vm    = (const float*)d_in[3];
    const float* Wm    = (const float*)d_in[4];
    const float* ln2_w = (const float*)d_in[5];
    const float* ln2_b = (const float*)d_in[6];
    const float* W1    = (const float*)d_in[7];
    const float* b1    = (const float*)d_in[8];
    const float* W2    = (const float*)d_in[9];
    const float* b2    = (const float*)d_in[10];
    float* out = (float*)d_out;

    // workspace layout (floats): sumsq[512] | h[B*N*16] | hw[B*N*16] | vxs[B*N*16]
    float* ws      = (float*)d_ws;
    float* sumsq   = ws;
    float* hbuf    = ws + BATCH * DDIM;
    float* hwbuf   = hbuf + (size_t)BATCH * NTOK * DDIM;
    float* vxsbuf  = hwbuf + (size_t)BATCH * NTOK * DDIM;
    (void)in_sizes; (void)n_in; (void)out_size; (void)ws_size;

    qab_zero_kernel<<<(BATCH * DDIM + 255) / 256, 256, 0, stream>>>(sumsq, BATCH * DDIM);

    dim3 g1(NTOK / 256, BATCH);
    qab_ln1_kernel<<<g1, 256, 0, stream>>>(x, ln1_w, ln1_b, hbuf, sumsq);

    qab_proj_kernel<<<(BATCH * NTOK) / 256, 256, 0, stream>>>(Vm, Wm, sumsq,
                                                              hbuf, hwbuf, vxsbuf);

    const size_t lds_bytes = (size_t)(16 * SROW + 16) * sizeof(float); // ~65.9KB of 320KB
    qab_attn_kernel<<<BATCH * (NTOK / 16), 32, lds_bytes, stream>>>(x, hbuf, hwbuf,
                                                                    vxsbuf, out);

    qab_mlp_kernel<<<(BATCH * NTOK) / 256, 256, 0, stream>>>(ln2_w, ln2_b,
                                                             W1, b1, W2, b2, out);
}